// Net_30425548325001
// MI455X (gfx1250) — compile-verified
//
#include <hip/hip_runtime.h>

// GCN: 2x GCNConv (linear -> normalized scatter-add -> bias -> relu) -> linear.
// All GEMMs use CDNA5 f32 WMMA (V_WMMA_F32_16X16X4_F32), one wave per 16x16 tile.

#define NN 100000     // nodes (multiple of 16)
#define NE 1600000    // edges
#define INF 128       // in features
#define HID 64        // hidden
#define NCL 16        // classes

typedef float v2f __attribute__((ext_vector_type(2)));
typedef float v8f __attribute__((ext_vector_type(8)));

// ---------- degree / normalization ----------

__global__ void init_deg(float* __restrict__ deg, int n) {
    int i = blockIdx.x * blockDim.x + threadIdx.x;
    if (i < n) deg[i] = 1.0f;   // self-loop contributes 1
}

__global__ void count_deg(const int* __restrict__ dst, float* __restrict__ deg, int e) {
    int i = blockIdx.x * blockDim.x + threadIdx.x;
    if (i < e) atomicAdd(&deg[dst[i]], 1.0f);
}

__global__ void deg_to_dinv(float* __restrict__ deg, int n) {
    int i = blockIdx.x * blockDim.x + threadIdx.x;
    if (i < n) deg[i] = rsqrtf(deg[i]);   // deg >= 1 always (self-loop)
}

__global__ void zero_buf(float* __restrict__ p, int n) {
    int i = blockIdx.x * blockDim.x + threadIdx.x;
    if (i < n) p[i] = 0.0f;
}

// ---------- WMMA GEMM: C[M x Ncols] = A[M x K] @ W[K x Ncols] (+bias) ----------
// One wave computes one 16x16 tile. M, Ncols multiples of 16; K multiple of 4.
// f32 WMMA layouts (ISA 7.12.2):
//   A 16x4 : lane l -> row m = l&15 ; VGPR v -> k = v + 2*(l>>4)
//   B 4x16 : lane l -> col n = l&15 ; VGPR v -> k = v + 2*(l>>4)
//   C 16x16: lane l, VGPR v -> m = v + 8*(l>>4), n = l&15
__global__ void gemm_wmma_f32(const float* __restrict__ A, const float* __restrict__ W,
                              const float* __restrict__ bias, float* __restrict__ C,
                              int M, int K, int Ncols, int addbias) {
    const int lane  = threadIdx.x & 31;
    const int wave  = threadIdx.x >> 5;
    const int ntile = Ncols >> 4;
    const int mtile = M >> 4;
    const int tile  = blockIdx.x * (blockDim.x >> 5) + wave;
    if (tile >= mtile * ntile) return;          // wave-uniform: EXEC stays all-ones
    const int mt = tile / ntile;
    const int nt = tile - mt * ntile;

    const int lm    = lane & 15;
    const int khalf = (lane >> 4) << 1;         // 0 or 2

    const float* arow = A + (size_t)(mt * 16 + lm) * K;        // row m, varying k
    const float* wcol = W + (size_t)(nt * 16 + lm);            // col n, varying k

    v8f acc = {0.f, 0.f, 0.f, 0.f, 0.f, 0.f, 0.f, 0.f};
    for (int k0 = 0; k0 < K; k0 += 4) {
        v2f a, b;
        a[0] = arow[k0 + khalf];
        a[1] = arow[k0 + khalf + 1];
        b[0] = wcol[(size_t)(k0 + khalf) * Ncols];
        b[1] = wcol[(size_t)(k0 + khalf + 1) * Ncols];
        // (neg_a, A, neg_b, B, c_mod, C, reuse_a, reuse_b)
        acc = __builtin_amdgcn_wmma_f32_16x16x4_f32(false, a, false, b,
                                                    (short)0, acc, false, false);
    }

    const float bv = addbias ? bias[nt * 16 + lm] : 0.0f;
    float* crow = C + (size_t)(mt * 16 + ((lane >> 4) << 3)) * Ncols + nt * 16 + lm;
#pragma unroll
    for (int v = 0; v < 8; ++v)
        crow[(size_t)v * Ncols] = acc[v] + bv;
}

// ---------- edge scatter: agg[dst] += dinv[src]*dinv[dst] * hw[src] ----------
// 16 threads per edge, 4 floats each (HID = 64). Working set is L2-resident.
__global__ void scatter_edges(const int* __restrict__ src, const int* __restrict__ dst,
                              const float* __restrict__ dinv, const float* __restrict__ hw,
                              float* __restrict__ agg, int e) {
    int t = blockIdx.x * blockDim.x + threadIdx.x;
    int eid = t >> 4;
    if (eid >= e) return;
    int f4 = (t & 15) << 2;
    int s = src[eid];
    int d = dst[eid];
    float w = dinv[s] * dinv[d];
    const float4 v = *(const float4*)(hw + (size_t)s * HID + f4);
    float* o = agg + (size_t)d * HID + f4;
    atomicAdd(o + 0, v.x * w);
    atomicAdd(o + 1, v.y * w);
    atomicAdd(o + 2, v.z * w);
    atomicAdd(o + 3, v.w * w);
}

// ---------- self-loop + bias + relu (in place on agg) ----------
__global__ void finalize_layer(const float* __restrict__ hw, const float* __restrict__ dinv,
                               const float* __restrict__ b, float* __restrict__ agg, int n) {
    int t = blockIdx.x * blockDim.x + threadIdx.x;
    int i = t / HID;
    if (i >= n) return;
    int f = t - i * HID;
    float di = dinv[i];
    float v = agg[t] + di * di * hw[t] + b[f];
    agg[t] = fmaxf(v, 0.0f);
}

static inline int cdiv(long long a, int b) { return (int)((a + b - 1) / b); }

extern "C" void kernel_launch(void* const* d_in, const int* in_sizes, int n_in,
                              void* d_out, int out_size, void* d_ws, size_t ws_size,
                              hipStream_t stream) {
    const float* x  = (const float*)d_in[0];
    const int*   ei = (const int*)d_in[1];   // [2, NE] flat: row0 = src, row1 = dst
    const float* W1 = (const float*)d_in[2];
    const float* b1 = (const float*)d_in[3];
    const float* W2 = (const float*)d_in[4];
    const float* b2 = (const float*)d_in[5];
    const float* Wl = (const float*)d_in[6];
    const float* bl = (const float*)d_in[7];
    float* out = (float*)d_out;

    float* dinv = (float*)d_ws;                       // NN floats
    float* buf1 = dinv + NN;                          // NN*HID floats (hw)
    float* buf2 = buf1 + (size_t)NN * HID;            // NN*HID floats (agg/h)

    const int* src = ei;
    const int* dst = ei + NE;

    const int B = 256;

    // normalization: deg -> dinv
    init_deg<<<cdiv(NN, B), B, 0, stream>>>(dinv, NN);
    count_deg<<<cdiv(NE, B), B, 0, stream>>>(dst, dinv, NE);
    deg_to_dinv<<<cdiv(NN, B), B, 0, stream>>>(dinv, NN);

    // ---- layer 1: hw = x @ W1 ; agg = Ahat * hw ; h1 = relu(agg + b1) ----
    {
        int tiles = (NN / 16) * (HID / 16);           // 25000
        gemm_wmma_f32<<<cdiv(tiles, 4), 128, 0, stream>>>(x, W1, nullptr, buf1,
                                                          NN, INF, HID, 0);
    }
    zero_buf<<<cdiv((long long)NN * HID, B), B, 0, stream>>>(buf2, NN * HID);
    scatter_edges<<<cdiv((long long)NE * 16, B), B, 0, stream>>>(src, dst, dinv, buf1, buf2, NE);
    finalize_layer<<<cdiv((long long)NN * HID, B), B, 0, stream>>>(buf1, dinv, b1, buf2, NN);

    // ---- layer 2: hw2 = h1 @ W2 ; agg ; h2 = relu(agg + b2) ----
    {
        int tiles = (NN / 16) * (HID / 16);
        gemm_wmma_f32<<<cdiv(tiles, 4), 128, 0, stream>>>(buf2, W2, nullptr, buf1,
                                                          NN, HID, HID, 0);
    }
    zero_buf<<<cdiv((long long)NN * HID, B), B, 0, stream>>>(buf2, NN * HID);
    scatter_edges<<<cdiv((long long)NE * 16, B), B, 0, stream>>>(src, dst, dinv, buf1, buf2, NE);
    finalize_layer<<<cdiv((long long)NN * HID, B), B, 0, stream>>>(buf1, dinv, b2, buf2, NN);

    // ---- classifier: out = h2 @ Wl + bl ----
    {
        int tiles = (NN / 16) * (NCL / 16);           // 6250
        gemm_wmma_f32<<<cdiv(tiles, 4), 128, 0, stream>>>(buf2, Wl, bl, out,
                                                          NN, HID, NCL, 1);
    }
}